// GraphSageLayer_27324581937607
// MI455X (gfx1250) — compile-verified
//
#include <hip/hip_runtime.h>

typedef __attribute__((ext_vector_type(2))) float v2f;
typedef __attribute__((ext_vector_type(8))) float v8f;

#define TPB           256     // 8 wave32
#define ROWS_PER_BLK  128     // 16 batch rows per wave
#define FDIM          128     // D
#define K2            256     // 2*D (concat)
#define S_NEIGH       10
#define C_STRIDE      260     // padded combined row stride (floats): 260 % 64 == 4 -> conflict-free A b64 loads
#define W_STRIDE      144     // padded weight row stride (floats): 2*144 % 64 == 32 -> half-waves 32 banks apart
#define SMEM_FLOATS   (K2 * W_STRIDE + ROWS_PER_BLK * C_STRIDE)   // 36864 + 33280 = 70144 floats = 280576 B < 320 KB

__global__ __launch_bounds__(TPB)
void sage_wmma_f32_kernel(const int* __restrict__ nodes,
                          const int* __restrict__ neigh,
                          const float* __restrict__ feat,
                          const float* __restrict__ weight,
                          float* __restrict__ out,
                          int Btot)
{
    extern __shared__ float smem[];
    float* sW = smem;                       // [256][W_STRIDE] weight rows (row-major, padded)
    float* sC = smem + K2 * W_STRIDE;       // [128][C_STRIDE] combined [self | mean] rows

    const int t    = threadIdx.x;
    const int lane = t & 31;
    const int wv   = t >> 5;                // wave id 0..7
    const int m    = lane & 15;             // position within half-wave
    const int hl   = lane >> 4;             // half-wave select
    const int blockBase = blockIdx.x * ROWS_PER_BLK;

    // ---- Phase 0a: stage weight into LDS (one 128-float row per thread) ----
    {
        const float4* src = (const float4*)(weight + (size_t)t * FDIM);
        float* dst = sW + t * W_STRIDE;
        #pragma unroll
        for (int i = 0; i < FDIM / 4; ++i) {
            float4 v = src[i];
            *(float4*)(dst + i * 4) = v;
        }
    }

    // ---- Phase 0b: gather self + mean(neighbors) into LDS combined rows ----
    {
        const int rowTile = blockBase + wv * 16;
        for (int r = 0; r < 16; ++r) {
            int b  = rowTile + r;
            int bc = (b < Btot) ? b : 0;            // clamp OOB reads; store is masked later
            float* crow = sC + (wv * 16 + r) * C_STRIDE;

            int node = nodes[bc];                    // wave-uniform -> scalar load
            float4 sf = ((const float4*)(feat + (size_t)node * FDIM))[lane];
            *(float4*)(crow + lane * 4) = sf;        // self features [0:128)

            float ax = 0.f, ay = 0.f, az = 0.f, aw = 0.f;
            #pragma unroll
            for (int s = 0; s < S_NEIGH; ++s) {
                int ni = neigh[bc * S_NEIGH + s];    // wave-uniform
                float4 nv = ((const float4*)(feat + (size_t)ni * FDIM))[lane];
                ax += nv.x; ay += nv.y; az += nv.z; aw += nv.w;
            }
            const float inv = 1.0f / (float)S_NEIGH;
            float4 mv; mv.x = ax * inv; mv.y = ay * inv; mv.z = az * inv; mv.w = aw * inv;
            *(float4*)(crow + FDIM + lane * 4) = mv; // mean features [128:256)
        }
    }
    __syncthreads();

    // ---- Phase 1: [16 x 256] x [256 x 128] via V_WMMA_F32_16X16X4_F32 ----
    // A 16x4 fp32 fragment: lane holds A[M=m][K = 4k + 2*hl + {0,1}]  (v2f, ds_load_b64)
    // B 4x16 fp32 fragment: lane holds W[K = 4k + 2*hl + {0,1}][N = nt*16 + m]
    v8f acc[8] = {};   // 8 N-tiles of 16x16 f32 accumulators (64 VGPRs)

    const float* aRow  = sC + (wv * 16 + m) * C_STRIDE + 2 * hl;
    const float* bBase = sW + (2 * hl) * W_STRIDE + m;

    for (int k = 0; k < K2 / 4; ++k) {
        v2f a = *(const v2f*)(aRow + 4 * k);                 // 8B-aligned -> ds_load_b64
        const float* bp = bBase + k * (4 * W_STRIDE);
        #pragma unroll
        for (int nt = 0; nt < 8; ++nt) {
            v2f bf;
            bf.x = bp[nt * 16];
            bf.y = bp[nt * 16 + W_STRIDE];
            acc[nt] = __builtin_amdgcn_wmma_f32_16x16x4_f32(
                /*neg_a=*/false, a, /*neg_b=*/false, bf,
                /*c_mod=*/(short)0, acc[nt],
                /*reuse_a=*/false, /*reuse_b=*/false);
        }
    }

    // ---- Phase 2: ReLU + store. D layout: component c -> row (c + 8*hl), col m ----
    const int rowTile = blockBase + wv * 16;
    #pragma unroll
    for (int c = 0; c < 8; ++c) {
        int b = rowTile + c + 8 * hl;
        if (b < Btot) {
            float* orow = out + (size_t)b * FDIM + m;
            #pragma unroll
            for (int nt = 0; nt < 8; ++nt) {
                orow[nt * 16] = fmaxf(acc[nt][c], 0.0f);
            }
        }
    }
}

extern "C" void kernel_launch(void* const* d_in, const int* in_sizes, int n_in,
                              void* d_out, int out_size, void* d_ws, size_t ws_size,
                              hipStream_t stream) {
    (void)n_in; (void)out_size; (void)d_ws; (void)ws_size;
    const int*   nodes  = (const int*)d_in[0];     // [B] int32
    const int*   neigh  = (const int*)d_in[1];     // [B, 10] int32
    const float* feat   = (const float*)d_in[2];   // [100000, 128] f32
    const float* weight = (const float*)d_in[3];   // [256, 128] f32
    float*       out    = (float*)d_out;           // [B, 128] f32

    int Btot = in_sizes[0];
    size_t smem = (size_t)SMEM_FLOATS * sizeof(float);

    // Allow >64KB dynamic LDS (gfx1250 WGP: 320KB). Deterministic, capture-safe.
    (void)hipFuncSetAttribute((const void*)sage_wmma_f32_kernel,
                              hipFuncAttributeMaxDynamicSharedMemorySize, (int)smem);

    int grid = (Btot + ROWS_PER_BLK - 1) / ROWS_PER_BLK;
    sage_wmma_f32_kernel<<<grid, TPB, smem, stream>>>(nodes, neigh, feat, weight, out, Btot);
}